// QuantumCircuitFeatures_61753039781982
// MI455X (gfx1250) — compile-verified
//
#include <hip/hip_runtime.h>
#include <math.h>

// ---------------- problem constants ----------------
#define HW    65536      // 256*256
#define OUTC  128
#define C2    256        // 2*OUT_C bn channels
#define EPSV  1e-5f

// ws layout (in floats)
#define OFF_PART 0       // 256 blocks * 5 partial sums
#define OFF_AB   1280    // A2[256], B2[256]
#define OFF_CIRC 2048    // 128 channels * 80 floats (v0:8, M1:32, H:32, pad:8)

typedef __attribute__((ext_vector_type(2))) float v2f;
typedef __attribute__((ext_vector_type(8))) float v8f;

struct cplx { float re, im; };
__device__ __forceinline__ cplx cmul(cplx a, cplx b) {
    return { a.re * b.re - a.im * b.im, a.re * b.im + a.im * b.re };
}
__device__ __forceinline__ cplx cadd(cplx a, cplx b) { return { a.re + b.re, a.im + b.im }; }
__device__ __forceinline__ cplx conjc(cplx a) { return { a.re, -a.im }; }

// ---------------- kernel 1: global pixel statistics ----------------
// Only 5 scalars needed: S0=sum c0, S1=sum c1, S00=sum c0^2, S01=sum c0*c1, S11=sum c1^2
__global__ void k_stats(const float* __restrict__ coords, float* __restrict__ part) {
    __shared__ float red[5][256];
    int p = blockIdx.x * 256 + threadIdx.x;
    float c0 = coords[p];
    float c1 = coords[HW + p];
    red[0][threadIdx.x] = c0;
    red[1][threadIdx.x] = c1;
    red[2][threadIdx.x] = c0 * c0;
    red[3][threadIdx.x] = c0 * c1;
    red[4][threadIdx.x] = c1 * c1;
    __syncthreads();
    for (int s = 128; s > 0; s >>= 1) {
        if (threadIdx.x < s) {
            #pragma unroll
            for (int k = 0; k < 5; k++) red[k][threadIdx.x] += red[k][threadIdx.x + s];
        }
        __syncthreads();
    }
    if (threadIdx.x == 0) {
        #pragma unroll
        for (int k = 0; k < 5; k++) part[blockIdx.x * 5 + k] = red[k][0];
    }
}

// ---------------- kernel 2: finalize BN -> per-channel affine (0.5 folded in) ----------------
__global__ void k_finalize(const float* __restrict__ part, const float* __restrict__ F,
                           const float* __restrict__ bnw, const float* __restrict__ bnb,
                           float* __restrict__ ab) {
    __shared__ float red[5][256];
    int t = threadIdx.x;
    #pragma unroll
    for (int k = 0; k < 5; k++) red[k][t] = part[t * 5 + k];
    __syncthreads();
    for (int s = 128; s > 0; s >>= 1) {
        if (t < s) {
            #pragma unroll
            for (int k = 0; k < 5; k++) red[k][t] += red[k][t + s];
        }
        __syncthreads();
    }
    float S0 = red[0][0], S1 = red[1][0], S00 = red[2][0], S01 = red[3][0], S11 = red[4][0];
    const float inv = 1.0f / (float)HW;
    float f0 = F[t * 2 + 0], f1 = F[t * 2 + 1];
    float mean = (f0 * S0 + f1 * S1) * inv;
    float e2   = (f0 * f0 * S00 + 2.0f * f0 * f1 * S01 + f1 * f1 * S11) * inv;
    float var  = e2 - mean * mean;                 // biased var, matches jnp.var
    float scale = bnw[t] * rsqrtf(var + EPSV);
    // angle = 0.5 * (scale*pre + (bias - mean*scale))
    ab[t]       = 0.5f * scale;
    ab[C2 + t]  = 0.5f * (bnb[t] - mean * scale);
}

// ---------------- kernel 3: per-channel circuit precompute (precise math, runs once) -------
__device__ __forceinline__ void u3mat(const float* __restrict__ p, cplx U[2][2]) {
    float st, ct; sincosf(p[0] * 0.5f, &st, &ct);
    float sl, cl; sincosf(p[2], &sl, &cl);
    float sp, cp; sincosf(p[1], &sp, &cp);
    float spl, cpl; sincosf(p[1] + p[2], &spl, &cpl);
    U[0][0] = { ct, 0.0f };
    U[0][1] = { -cl * st, -sl * st };
    U[1][0] = { cp * st,  sp * st };
    U[1][1] = { cpl * ct, spl * ct };
}
__device__ __forceinline__ void apply_w0(cplx v[4], const cplx U[2][2]) {
    cplx n0 = cadd(cmul(U[0][0], v[0]), cmul(U[0][1], v[2]));
    cplx n1 = cadd(cmul(U[0][0], v[1]), cmul(U[0][1], v[3]));
    cplx n2 = cadd(cmul(U[1][0], v[0]), cmul(U[1][1], v[2]));
    cplx n3 = cadd(cmul(U[1][0], v[1]), cmul(U[1][1], v[3]));
    v[0] = n0; v[1] = n1; v[2] = n2; v[3] = n3;
}
__device__ __forceinline__ void apply_w1(cplx v[4], const cplx U[2][2]) {
    cplx n0 = cadd(cmul(U[0][0], v[0]), cmul(U[0][1], v[1]));
    cplx n1 = cadd(cmul(U[1][0], v[0]), cmul(U[1][1], v[1]));
    cplx n2 = cadd(cmul(U[0][0], v[2]), cmul(U[0][1], v[3]));
    cplx n3 = cadd(cmul(U[1][0], v[2]), cmul(U[1][1], v[3]));
    v[0] = n0; v[1] = n1; v[2] = n2; v[3] = n3;
}
__device__ __forceinline__ void apply_cnot(cplx v[4]) { cplx t = v[2]; v[2] = v[3]; v[3] = t; }

__device__ __forceinline__ void mat_layer(cplx M[4][4], const cplx U0[2][2], const cplx U1[2][2]) {
    #pragma unroll
    for (int c = 0; c < 4; c++) {
        cplx v[4] = { M[0][c], M[1][c], M[2][c], M[3][c] };
        apply_w0(v, U0); apply_w1(v, U1); apply_cnot(v);
        M[0][c] = v[0]; M[1][c] = v[1]; M[2][c] = v[2]; M[3][c] = v[3];
    }
}

__global__ void k_circuit(const float* __restrict__ q, float* __restrict__ circ) {
    int o = threadIdx.x;
    if (o >= OUTC) return;
    // q flat layout: (3, 2, 256, 3) -> idx = ((i*2+j)*256 + (2*o+wire))*3
    cplx psi[4] = { {1.f, 0.f}, {0.f, 0.f}, {0.f, 0.f}, {0.f, 0.f} };
    #pragma unroll
    for (int j = 0; j < 2; j++) {
        cplx U0[2][2], U1[2][2];
        u3mat(&q[(((0 * 2 + j) * C2) + (2 * o + 0)) * 3], U0);
        u3mat(&q[(((0 * 2 + j) * C2) + (2 * o + 1)) * 3], U1);
        apply_w0(psi, U0); apply_w1(psi, U1); apply_cnot(psi);
    }
    cplx M1[4][4], M2[4][4];
    #pragma unroll
    for (int r = 0; r < 4; r++)
        #pragma unroll
        for (int c = 0; c < 4; c++) {
            M1[r][c] = { (r == c) ? 1.f : 0.f, 0.f };
            M2[r][c] = { (r == c) ? 1.f : 0.f, 0.f };
        }
    #pragma unroll
    for (int j = 0; j < 2; j++) {
        cplx U0[2][2], U1[2][2];
        u3mat(&q[(((1 * 2 + j) * C2) + (2 * o + 0)) * 3], U0);
        u3mat(&q[(((1 * 2 + j) * C2) + (2 * o + 1)) * 3], U1);
        mat_layer(M1, U0, U1);
        u3mat(&q[(((2 * 2 + j) * C2) + (2 * o + 0)) * 3], U0);
        u3mat(&q[(((2 * 2 + j) * C2) + (2 * o + 1)) * 3], U1);
        mat_layer(M2, U0, U1);
    }
    // H = M2^dagger * diag(1,1,-1,-1) * M2  (Hermitian)
    float* base = circ + o * 80;
    #pragma unroll
    for (int k = 0; k < 4; k++) { base[2 * k] = psi[k].re; base[2 * k + 1] = psi[k].im; }
    #pragma unroll
    for (int j = 0; j < 4; j++)
        #pragma unroll
        for (int k = 0; k < 4; k++) {
            base[8 + (j * 4 + k) * 2]     = M1[j][k].re;
            base[8 + (j * 4 + k) * 2 + 1] = M1[j][k].im;
            cplx h = { 0.f, 0.f };
            #pragma unroll
            for (int r = 0; r < 4; r++) {
                float g = (r < 2) ? 1.f : -1.f;
                cplx tt = cmul(conjc(M2[r][j]), M2[r][k]);
                h.re += g * tt.re; h.im += g * tt.im;
            }
            base[40 + (j * 4 + k) * 2]     = h.re;
            base[40 + (j * 4 + k) * 2 + 1] = h.im;
        }
}

// ---------------- per-pixel circuit evaluation (hardware trig) ----------------
__device__ __forceinline__ float eval_circ(const float* __restrict__ cc, float alpha, float beta) {
    // fast hardware transcendentals: angles are BN-normalized (small), v_sin/v_cos accuracy is ample
    float sa = __sinf(alpha), ca = __cosf(alpha);
    float sb = __sinf(beta),  cb = __cosf(beta);
    float rr = ca * cb, ii = sa * sb, ri = ca * sb, ir = sa * cb;
    cplx d3 = { rr - ii, ir + ri };   // e^{i(a+b)}
    cplx d2 = { rr + ii, ir - ri };   // e^{i(a-b)}
    cplx d[4];
    d[0] = conjc(d3); d[1] = conjc(d2); d[2] = d2; d[3] = d3;   // basis k = 2*w0 + w1
    cplx u[4];
    #pragma unroll
    for (int k = 0; k < 4; k++) {
        cplx v0k = { cc[2 * k], cc[2 * k + 1] };
        u[k] = cmul(d[k], v0k);
    }
    cplx w[4];
    #pragma unroll
    for (int j = 0; j < 4; j++) {
        cplx tj = { 0.f, 0.f };
        #pragma unroll
        for (int k = 0; k < 4; k++) {
            cplx m = { cc[8 + (j * 4 + k) * 2], cc[8 + (j * 4 + k) * 2 + 1] };
            tj = cadd(tj, cmul(m, u[k]));
        }
        w[j] = cmul(d[j], tj);
    }
    // ev = w^dag H w with Hermitian H: diagonal + 2*Re(upper triangle)
    float ev = 0.f;
    #pragma unroll
    for (int j = 0; j < 4; j++) {
        float hd = cc[40 + (j * 4 + j) * 2];
        ev += hd * (w[j].re * w[j].re + w[j].im * w[j].im);
    }
    #pragma unroll
    for (int j = 0; j < 4; j++)
        #pragma unroll
        for (int k = j + 1; k < 4; k++) {
            cplx h = { cc[40 + (j * 4 + k) * 2], cc[40 + (j * 4 + k) * 2 + 1] };
            cplx z = cmul(conjc(w[j]), w[k]);
            ev += 2.0f * (h.re * z.re - h.im * z.im);
        }
    return ev;
}

// ---------------- kernel 4: main fused kernel ----------------
// grid = (512, 16), block = 256 (8 waves). Wave w: 16 pixels starting at (blockIdx.x*8+w)*16.
// blockIdx.y selects 16 bn channels (= 8 output channels). One V_WMMA_F32_16X16X4_F32 computes
// the 16x16 pre-activation tile; each lane then owns 1 pixel x 4 output channels.
__global__ void __launch_bounds__(256) k_main(
    const float* __restrict__ coords, const float* __restrict__ F,
    const float* __restrict__ ab, const float* __restrict__ circ,
    float* __restrict__ out)
{
    __shared__ float lds_circ[8 * 80];
    __shared__ float lds_ab[32];
    const int t = threadIdx.x;
    const int m0   = blockIdx.y * 16;   // bn-channel tile base
    const int och0 = blockIdx.y * 8;    // output-channel tile base

    for (int i = t; i < 8 * 80; i += 256) lds_circ[i] = circ[och0 * 80 + i];
    if (t < 32) lds_ab[t] = (t < 16) ? ab[m0 + t] : ab[C2 + m0 + (t - 16)];
    __syncthreads();

    const int wv    = t >> 5;
    const int lane  = t & 31;
    const int lhalf = lane >> 4;        // 0: channels m0..m0+7, 1: m0+8..m0+15
    const int l16   = lane & 15;
    const int p0 = (blockIdx.x * 8 + wv) * 16;
    const bool lo = (lane < 16);

    // Branchless operand build: all lanes load (lane&15) address (coalesced),
    // lanes 16-31 (K=2,3 padding) are zeroed with a select — EXEC stays full for WMMA.
    float fa0 = F[(m0 + l16) * 2 + 0];
    float fa1 = F[(m0 + l16) * 2 + 1];
    float cc0 = coords[p0 + l16];
    float cc1 = coords[HW + p0 + l16];
    v2f a, b;
    a.x = lo ? fa0 : 0.0f;  a.y = lo ? fa1 : 0.0f;
    b.x = lo ? cc0 : 0.0f;  b.y = lo ? cc1 : 0.0f;

    v8f acc = { 0.f, 0.f, 0.f, 0.f, 0.f, 0.f, 0.f, 0.f };
    acc = __builtin_amdgcn_wmma_f32_16x16x4_f32(false, a, false, b, (short)0, acc,
                                                false, false);
    // acc[r] in lane L: pre[m0 + lhalf*8 + r, p0 + (L&15)]

    const int p = p0 + l16;
    #pragma unroll
    for (int c = 0; c < 4; c++) {
        float preE = acc[2 * c];
        float preO = acc[2 * c + 1];
        int mloc = lhalf * 8 + 2 * c;               // local bn channel (even of pair)
        float alpha = lds_ab[mloc]     * preE + lds_ab[16 + mloc];
        float beta  = lds_ab[mloc + 1] * preO + lds_ab[16 + mloc + 1];
        int ochLoc = lhalf * 4 + c;
        float ev = eval_circ(&lds_circ[ochLoc * 80], alpha, beta);
        size_t o0 = ((size_t)(och0 + ochLoc) << 16) + (size_t)p;
        out[o0]                    = ev;   // batch 0
        out[o0 + (size_t)128 * HW] = ev;   // batch 1 (output tiled over batch)
        out[o0 + (size_t)256 * HW] = ev;   // batch 2
        out[o0 + (size_t)384 * HW] = ev;   // batch 3
    }
}

// ---------------- launcher ----------------
extern "C" void kernel_launch(void* const* d_in, const int* in_sizes, int n_in,
                              void* d_out, int out_size, void* d_ws, size_t ws_size,
                              hipStream_t stream) {
    const float* coords = (const float*)d_in[0];   // (4,2,256,256) f32 (batch 0 used)
    const float* F      = (const float*)d_in[1];   // (256,2) f32
    const float* q      = (const float*)d_in[2];   // (1,3,2,256,3) f32
    const float* bnw    = (const float*)d_in[3];   // (256,) f32
    const float* bnb    = (const float*)d_in[4];   // (256,) f32
    float* out = (float*)d_out;                    // (4,128,256,256) f32
    float* ws  = (float*)d_ws;

    float* part = ws + OFF_PART;
    float* ab   = ws + OFF_AB;
    float* circ = ws + OFF_CIRC;

    k_stats   <<<256, 256, 0, stream>>>(coords, part);
    k_finalize<<<1,   256, 0, stream>>>(part, F, bnw, bnb, ab);
    k_circuit <<<1,   128, 0, stream>>>(q, circ);
    dim3 grid(512, 16);
    k_main    <<<grid, 256, 0, stream>>>(coords, F, ab, circ, out);
}